// OSSM_12137577579132
// MI455X (gfx1250) — compile-verified
//
#include <hip/hip_runtime.h>
#include <math.h>

typedef __attribute__((ext_vector_type(2))) float v2f;
typedef __attribute__((ext_vector_type(8))) float v8f;

// Problem constants (B=1)
constexpr int Dch  = 96;
constexpr int Hh   = 64;
constexpr int Ww   = 64;
constexpr int Ll   = Hh * Ww;        // 4096
constexpr int Kk   = 8;
constexpr int Nn   = 16;
constexpr int Rr   = 6;
constexpr int CDBL = Rr + 2 * Nn;    // 38 valid rows of x_dbl
constexpr int CPAD = 48;             // padded to 3 full 16-row WMMA tiles
constexpr int KD   = Kk * Dch;       // 768
constexpr int NCH  = 8;              // scan chunks
constexpr int CHL  = Ll / NCH;       // 512
constexpr int MT   = CPAD / 16;      // 3 row tiles
constexpr int KS   = Dch / 4;        // 24 WMMA K-steps
constexpr int NTW  = 4;              // column tiles per wave (A-fragment reuse)

// ---------------------------------------------------------------------------
// 0) zero-pad x_proj_weight rows 38..47 so the GEMM needs no masking at all
// ---------------------------------------------------------------------------
__global__ void k_pad_w(const float* __restrict__ Wp, float* __restrict__ Wpad) {
    int idx = blockIdx.x * blockDim.x + threadIdx.x;   // over Kk*CPAD*Dch
    if (idx >= Kk * CPAD * Dch) return;
    int d = idx % Dch;
    int c = (idx / Dch) % CPAD;
    int k = idx / (Dch * CPAD);
    Wpad[idx] = (c < CDBL) ? Wp[(k * CDBL + c) * Dch + d] : 0.f;
}

// ---------------------------------------------------------------------------
// 1) cross_scan: xs[k,d,l] = x[d, h(k,l), w(k,l)]  (8 directional views)
// ---------------------------------------------------------------------------
__global__ void k_cross_scan(const float* __restrict__ x, float* __restrict__ xs) {
    int idx = blockIdx.x * blockDim.x + threadIdx.x;
    if (idx >= Kk * Dch * Ll) return;
    int l = idx % Ll;
    int d = (idx / Ll) % Dch;
    int k = idx / (Ll * Dch);
    bool rev = (k == 2) || (k == 3) || (k == 6) || (k == 7);
    int lr = rev ? (Ll - 1 - l) : l;
    int base = ((k & 4) >> 1) | (k & 1);   // 0:row 1:col 2:diag 3:antidiag
    int h, w;
    switch (base) {
        case 0:  h = lr >> 6; w = lr & 63; break;
        case 1:  h = lr & 63; w = lr >> 6; break;
        case 2: { int hq = lr & 63, wq = lr >> 6; h = hq; w = (hq + wq) & 63; } break;
        default:{ int hq = lr & 63, wq = lr >> 6; h = hq; w = (wq - hq + 64) & 63; } break;
    }
    xs[idx] = x[d * Ll + h * Ww + w];
}

// ---------------------------------------------------------------------------
// 2) x_dbl[k] = Wpad_k(48x96) @ xs_k(96x4096) via V_WMMA_F32_16X16X4_F32
//    A fragments preloaded once as b64 loads, reused for 4 col tiles;
//    per tile: batched B-load clause then 24 chained WMMAs; unconditional
//    stores into the 48-row padded x_dbl.
// ---------------------------------------------------------------------------
__global__ void k_gemm_xdbl(const float* __restrict__ Wpad, const float* __restrict__ xs,
                            float* __restrict__ xdbl) {
    int wave = (blockIdx.x * blockDim.x + threadIdx.x) >> 5;   // exact grid, all lanes live
    int lane = threadIdx.x & 31;
    constexpr int NGRP = Ll / (16 * NTW);         // 64 column groups
    constexpr int TPK  = MT * NGRP;               // tiles per direction
    int k   = wave / TPK;
    int rem = wave % TPK;
    int mt  = rem / NGRP;                         // 0..2 row tile
    int ng  = rem % NGRP;                         // 0..63 column group
    int c0 = mt * 16;
    int lbase = ng * (16 * NTW);
    int m = lane & 15, half = lane >> 4;

    const float* Wk = Wpad + k * CPAD * Dch;
    const float* Xk = xs + (size_t)k * Dch * Ll;

    // --- A fragment preload: VGPR v holds K = v + 2*half; pair is contiguous ---
    int c = c0 + m;
    v2f areg[KS];
#pragma unroll
    for (int s = 0; s < KS; ++s) {
        int ka = 4 * s + 2 * half;
        areg[s] = *(const v2f*)(Wk + c * Dch + ka);
    }

    float* Ok = xdbl + (size_t)k * CPAD * Ll;
#pragma unroll
    for (int t = 0; t < NTW; ++t) {
        int l0 = lbase + t * 16;
        v2f breg[KS];
#pragma unroll
        for (int s = 0; s < KS; ++s) {
            int ka = 4 * s + 2 * half;
            breg[s][0] = Xk[(size_t)ka       * Ll + l0 + m];
            breg[s][1] = Xk[(size_t)(ka + 1) * Ll + l0 + m];
        }
        v8f acc = {};
#pragma unroll
        for (int s = 0; s < KS; ++s)
            acc = __builtin_amdgcn_wmma_f32_16x16x4_f32(
                      false, areg[s], false, breg[s], (short)0, acc, false, false);
#pragma unroll
        for (int r = 0; r < 8; ++r) {
            int co = c0 + r + 8 * half;            // D layout: VGPR r -> M = r + 8*half
            Ok[(size_t)co * Ll + l0 + m] = acc[r]; // co < 48 always, no guard
        }
    }
}

// ---------------------------------------------------------------------------
// 3) delta[kd,l] = softplus( dt_bias[kd] + sum_r x_dbl[k,r,l] * dtw[kd,r] )
// ---------------------------------------------------------------------------
__global__ void k_delta(const float* __restrict__ xdbl, const float* __restrict__ dtw,
                        const float* __restrict__ dtb, float* __restrict__ delta) {
    int idx = blockIdx.x * blockDim.x + threadIdx.x;
    if (idx >= KD * Ll) return;
    int l  = idx % Ll;
    int kd = idx / Ll;
    int k  = kd / Dch;
    const float* xr = xdbl + (size_t)k * CPAD * Ll + l;
    const float* wr = dtw + kd * Rr;
    float s = dtb[kd];
#pragma unroll
    for (int r = 0; r < Rr; ++r) s = fmaf(xr[(size_t)r * Ll], wr[r], s);
    delta[idx] = (s > 20.f) ? s : log1pf(__expf(s));   // softplus
}

// ---------------------------------------------------------------------------
// 4a) chunk-local scan: per (kd, chunk, n) compute prod(dA) and final h (h0=0)
// ---------------------------------------------------------------------------
__global__ void k_scan_chunk(const float* __restrict__ xs, const float* __restrict__ delta,
                             const float* __restrict__ xdbl, const float* __restrict__ Alog,
                             float* __restrict__ cAprod, float* __restrict__ cH) {
    int g  = (blockIdx.x * blockDim.x + threadIdx.x) >> 4;  // channel-chunk group
    int n  = threadIdx.x & 15;                              // state index
    int kd = g / NCH, ch = g % NCH;
    int k  = kd / Dch;
    float An = -__expf(Alog[kd * Nn + n]);
    const float* Brow = xdbl + (size_t)k * CPAD * Ll + (size_t)(Rr + n) * Ll;
    const float* dl   = delta + (size_t)kd * Ll;
    const float* ur   = xs    + (size_t)kd * Ll;
    float aprod = 1.f, h = 0.f;
    int l0 = ch * CHL;
    for (int i = 0; i < CHL; ++i) {
        int l = l0 + i;
        float dt = dl[l];
        float da = __expf(dt * An);
        h = fmaf(da, h, dt * Brow[l] * ur[l]);
        aprod *= da;
    }
    cAprod[g * Nn + n] = aprod;
    cH[g * Nn + n]     = h;
}

// ---------------------------------------------------------------------------
// 4b) serial carry across the 8 chunks of each (kd, n)
// ---------------------------------------------------------------------------
__global__ void k_scan_carry(const float* __restrict__ cAprod, const float* __restrict__ cH,
                             float* __restrict__ carry) {
    int tid = blockIdx.x * blockDim.x + threadIdx.x;
    if (tid >= KD * Nn) return;
    int kd = tid / Nn, n = tid % Nn;
    float hc = 0.f;
    for (int ch = 0; ch < NCH; ++ch) {
        int g = kd * NCH + ch;
        carry[g * Nn + n] = hc;                       // state entering chunk ch
        hc = fmaf(cAprod[g * Nn + n], hc, cH[g * Nn + n]);
    }
}

// ---------------------------------------------------------------------------
// 4c) re-scan with corrected initial state, emit ys = sum_n C_n h_n + D*u
//     16 lanes (= states) per channel-chunk; width-16 xor-shuffle reduction
// ---------------------------------------------------------------------------
__global__ void k_scan_apply(const float* __restrict__ xs, const float* __restrict__ delta,
                             const float* __restrict__ xdbl, const float* __restrict__ Alog,
                             const float* __restrict__ Ds, const float* __restrict__ carry,
                             float* __restrict__ ys) {
    int g  = (blockIdx.x * blockDim.x + threadIdx.x) >> 4;
    int n  = threadIdx.x & 15;
    int kd = g / NCH, ch = g % NCH;
    int k  = kd / Dch;
    float An = -__expf(Alog[kd * Nn + n]);
    const float* Brow = xdbl + (size_t)k * CPAD * Ll + (size_t)(Rr + n) * Ll;
    const float* Crow = xdbl + (size_t)k * CPAD * Ll + (size_t)(Rr + Nn + n) * Ll;
    const float* dl   = delta + (size_t)kd * Ll;
    const float* ur   = xs    + (size_t)kd * Ll;
    float Dv = Ds[kd];
    float h  = carry[g * Nn + n];
    int l0 = ch * CHL;
    for (int i = 0; i < CHL; ++i) {
        int l = l0 + i;
        float dt = dl[l], u = ur[l];
        float da = __expf(dt * An);
        h = fmaf(da, h, dt * Brow[l] * u);
        float p = Crow[l] * h;
        p += __shfl_xor(p, 8, 16);
        p += __shfl_xor(p, 4, 16);
        p += __shfl_xor(p, 2, 16);
        p += __shfl_xor(p, 1, 16);
        if (n == 0) ys[(size_t)kd * Ll + l] = fmaf(Dv, u, p);
    }
}

// ---------------------------------------------------------------------------
// 5) cross_merge (inverse gathers of the 8 directions) + channel LayerNorm
//    one wave per spatial position l; 3 channels per lane (96 = 3*32)
// ---------------------------------------------------------------------------
__global__ void k_merge_ln(const float* __restrict__ ys, const float* __restrict__ gamma,
                           const float* __restrict__ beta, float* __restrict__ out) {
    int wave = threadIdx.x >> 5;
    int lane = threadIdx.x & 31;
    int l = blockIdx.x * 8 + wave;
    int h = l >> 6, w = l & 63;
    int lt = (w << 6) + h;                          // transpose view
    int ld = (((w - h + 64) & 63) << 6) + h;        // diag-scatter source
    int la = (((w + h) & 63) << 6) + h;             // antidiag-scatter source
    const size_t DL = (size_t)Dch * Ll;

    float yv[3];
    float s = 0.f, sq = 0.f;
#pragma unroll
    for (int j = 0; j < 3; ++j) {
        int d = lane + 32 * j;
        const float* yd = ys + (size_t)d * Ll;
        float v = yd[0 * DL + l]  + yd[2 * DL + (Ll - 1 - l)]
                + yd[1 * DL + lt] + yd[3 * DL + (Ll - 1 - lt)]
                + yd[4 * DL + ld] + yd[6 * DL + (Ll - 1 - ld)]
                + yd[5 * DL + la] + yd[7 * DL + (Ll - 1 - la)];
        yv[j] = v; s += v; sq += v * v;
    }
#pragma unroll
    for (int off = 16; off > 0; off >>= 1) {
        s  += __shfl_xor(s,  off, 32);
        sq += __shfl_xor(sq, off, 32);
    }
    float mean = s * (1.f / Dch);
    float var  = sq * (1.f / Dch) - mean * mean;
    float rs   = rsqrtf(var + 1e-5f);
#pragma unroll
    for (int j = 0; j < 3; ++j) {
        int d = lane + 32 * j;
        out[(size_t)l * Dch + d] = (yv[j] - mean) * rs * gamma[d] + beta[d];
    }
}

// ---------------------------------------------------------------------------
extern "C" void kernel_launch(void* const* d_in, const int* in_sizes, int n_in,
                              void* d_out, int out_size, void* d_ws, size_t ws_size,
                              hipStream_t stream) {
    const float* x     = (const float*)d_in[0];   // [96,64,64]
    const float* xpw   = (const float*)d_in[1];   // [8,38,96]
    const float* dtw   = (const float*)d_in[2];   // [8,96,6]
    const float* dtb   = (const float*)d_in[3];   // [8,96]
    const float* Alog  = (const float*)d_in[4];   // [768,16]
    const float* Ds    = (const float*)d_in[5];   // [768]
    const float* gamma = (const float*)d_in[6];   // [96]
    const float* beta  = (const float*)d_in[7];   // [96]
    float* out = (float*)d_out;                   // [64,64,96]

    // workspace carve-up (floats; all offsets even -> b64-aligned)
    float* ws    = (float*)d_ws;
    float* xs    = ws;                               // 8*96*4096  = 3,145,728
    float* xdbl  = xs    + (size_t)Kk * Dch * Ll;    // 8*48*4096  = 1,572,864 (padded)
    float* delta = xdbl  + (size_t)Kk * CPAD * Ll;   // 3,145,728
    float* ysb   = delta + (size_t)KD * Ll;          // 3,145,728
    float* cA    = ysb   + (size_t)KD * Ll;          // 768*8*16   = 98,304
    float* cH    = cA    + (size_t)KD * NCH * Nn;    // 98,304
    float* carry = cH    + (size_t)KD * NCH * Nn;    // 98,304
    float* wpad  = carry + (size_t)KD * NCH * Nn;    // 8*48*96    = 36,864

    const int TB = 256;

    // 0) pad weights to 48 rows (kills all masking in the WMMA kernel)
    k_pad_w<<<(Kk * CPAD * Dch) / TB, TB, 0, stream>>>(xpw, wpad);

    // 1) cross_scan: 8*96*4096 elements
    k_cross_scan<<<(Kk * Dch * Ll) / TB, TB, 0, stream>>>(x, xs);

    // 2) WMMA GEMM: 8 dirs * 3 row-tiles * 64 col-groups = 1536 waves
    k_gemm_xdbl<<<(Kk * MT * (Ll / (16 * NTW)) * 32) / TB, TB, 0, stream>>>(wpad, xs, xdbl);

    // 3) dt projection + softplus: 768*4096 elements
    k_delta<<<(KD * Ll) / TB, TB, 0, stream>>>(xdbl, dtw, dtb, delta);

    // 4) chunked selective scan
    k_scan_chunk<<<(KD * NCH * Nn) / TB, TB, 0, stream>>>(xs, delta, xdbl, Alog, cA, cH);
    k_scan_carry<<<(KD * Nn + TB - 1) / TB, TB, 0, stream>>>(cA, cH, carry);
    k_scan_apply<<<(KD * NCH * Nn) / TB, TB, 0, stream>>>(xs, delta, xdbl, Alog, Ds, carry, ysb);

    // 5) cross_merge + LayerNorm: one wave per l, 8 waves/block
    k_merge_ln<<<Ll / 8, TB, 0, stream>>>(ysb, gamma, beta, out);
}